// SGC_Net_78030965834313
// MI455X (gfx1250) — compile-verified
//
#include <hip/hip_runtime.h>
#include <math.h>

#define N_NODES   170000
#define N_EDGES   1200000
#define D_FEAT    128
#define N_CLASSES 40
#define N_TILES   (N_NODES / 16)   // 10625, exact

typedef __attribute__((ext_vector_type(2))) float v2f;
typedef __attribute__((ext_vector_type(8))) float v8f;

// ---------------- degree / normalization ----------------

__global__ void k_deg_init(float* __restrict__ deg) {
    int i = blockIdx.x * blockDim.x + threadIdx.x;
    if (i < N_NODES) deg[i] = 1.0f;            // self-loop
}

__global__ void k_deg_accum(const long long* __restrict__ dst,
                            float* __restrict__ deg) {
    int e = blockIdx.x * blockDim.x + threadIdx.x;
    if (e < N_EDGES) atomicAdd(&deg[(int)dst[e]], 1.0f);
}

__global__ void k_dinv(float* __restrict__ deg) {
    int i = blockIdx.x * blockDim.x + threadIdx.x;
    if (i < N_NODES) deg[i] = rsqrtf(deg[i]);  // deg >= 1 always
}

// ---------------- propagation ----------------

// h_out[i] = dinv[i]^2 * h_in[i]   (self-loop term; also initializes h_out)
__global__ void k_self(const float* __restrict__ hin,
                       const float* __restrict__ dinv,
                       float* __restrict__ hout) {
    long long t = (long long)blockIdx.x * blockDim.x + threadIdx.x;
    if (t >= (long long)N_NODES * (D_FEAT / 4)) return;
    int node = (int)(t >> 5);
    int f4   = (int)(t & 31);
    float w = dinv[node]; w *= w;
    float4 v = ((const float4*)(hin + (size_t)node * D_FEAT))[f4];
    float4 o; o.x = w * v.x; o.y = w * v.y; o.z = w * v.z; o.w = w * v.w;
    ((float4*)(hout + (size_t)node * D_FEAT))[f4] = o;
}

// one wave per edge; each lane moves a float4 and scatter-adds it
__global__ __launch_bounds__(256)
void k_edge(const long long* __restrict__ src,
            const long long* __restrict__ dst,
            const float* __restrict__ dinv,
            const float* __restrict__ hin,
            float* __restrict__ hout) {
    int wave = (blockIdx.x * blockDim.x + threadIdx.x) >> 5;
    int lane = threadIdx.x & 31;
    if (wave >= N_EDGES) return;
    int s = (int)src[wave];
    int d = (int)dst[wave];
    float w = dinv[s] * dinv[d];
    float4 v = ((const float4*)(hin + (size_t)s * D_FEAT))[lane];
    float* o = hout + (size_t)d * D_FEAT + lane * 4;
    atomicAdd(o + 0, w * v.x);
    atomicAdd(o + 1, w * v.y);
    atomicAdd(o + 2, w * v.z);
    atomicAdd(o + 3, w * v.w);
}

// ---------------- WMMA classifier + fused log-softmax ----------------
// logits = h @ W^T + b, then log_softmax per row. One wave per 16-row tile.
__global__ __launch_bounds__(256)
void k_logits(const float* __restrict__ h,
              const float* __restrict__ W,
              const float* __restrict__ b,
              float* __restrict__ out) {
    // W transposed and K-paired: Wp[p][c] = {W[c][2p], W[c][2p+1]},
    // classes padded 40->48. Each B operand is one aligned ds_load_b64.
    __shared__ v2f Wp[D_FEAT / 2][48];         // 64*48*8 = 24576 bytes
    for (int idx = threadIdx.x; idx < (D_FEAT / 2) * 48; idx += blockDim.x) {
        int p = idx / 48, c = idx % 48;
        v2f w = {0.0f, 0.0f};
        if (c < N_CLASSES) {
            w.x = W[(size_t)c * D_FEAT + 2 * p];
            w.y = W[(size_t)c * D_FEAT + 2 * p + 1];
        }
        Wp[p][c] = w;
    }
    __syncthreads();

    int lane  = threadIdx.x & 31;
    int tile  = blockIdx.x * (blockDim.x >> 5) + (threadIdx.x >> 5);
    if (tile >= N_TILES) return;               // wave-uniform: EXEC stays all-1s

    int mbase = tile * 16;
    int mrow  = mbase + (lane & 15);           // A-matrix: both halves cover M=0..15
    int khalf = lane >> 4;                     // 0: K+{0,1}, 1: K+{2,3}
    int n     = lane & 15;

    const v2f* arow = (const v2f*)(h + (size_t)mrow * D_FEAT);

    v8f acc0 = {}, acc1 = {}, acc2 = {};
    #pragma unroll
    for (int k0 = 0; k0 < D_FEAT; k0 += 4) {
        int p = (k0 >> 1) + khalf;             // K pair index for this half
        v2f a  = arow[p];                      // A[m][k0+2*khalf .. +1]
        v2f b0 = Wp[p][n];
        v2f b1 = Wp[p][16 + n];
        v2f b2 = Wp[p][32 + n];
        acc0 = __builtin_amdgcn_wmma_f32_16x16x4_f32(false, a, false, b0, (short)0, acc0, false, false);
        acc1 = __builtin_amdgcn_wmma_f32_16x16x4_f32(false, a, false, b1, (short)0, acc1, false, false);
        acc2 = __builtin_amdgcn_wmma_f32_16x16x4_f32(false, a, false, b2, (short)0, acc2, false, false);
    }

    // bias; classes 32..47 valid only for n<8 (classes 32..39)
    bool  cvalid = (n < 8);
    float bb0 = b[n];
    float bb1 = b[16 + n];
    float bb2 = cvalid ? b[32 + n] : 0.0f;

    // C/D layout: VGPR j -> row = mbase + j + 8*khalf, col = n (+16*ctile)
    #pragma unroll
    for (int j = 0; j < 8; ++j) {
        int row = mbase + j + 8 * khalf;
        float v0 = acc0[j] + bb0;
        float v1 = acc1[j] + bb1;
        float v2 = cvalid ? (acc2[j] + bb2) : -INFINITY;

        float mx = fmaxf(fmaxf(v0, v1), v2);
        #pragma unroll
        for (int off = 1; off < 16; off <<= 1)
            mx = fmaxf(mx, __shfl_xor(mx, off, 32));   // stays within 16-lane half

        float s = expf(v0 - mx) + expf(v1 - mx) + (cvalid ? expf(v2 - mx) : 0.0f);
        #pragma unroll
        for (int off = 1; off < 16; off <<= 1)
            s += __shfl_xor(s, off, 32);

        float lse = mx + logf(s);
        float* orow = out + (size_t)row * N_CLASSES;
        orow[n]      = v0 - lse;
        orow[16 + n] = v1 - lse;
        if (cvalid) orow[32 + n] = v2 - lse;
    }
}

// ---------------- launch ----------------

extern "C" void kernel_launch(void* const* d_in, const int* in_sizes, int n_in,
                              void* d_out, int out_size, void* d_ws, size_t ws_size,
                              hipStream_t stream) {
    const float*     x   = (const float*)d_in[0];
    const long long* ei  = (const long long*)d_in[1];
    const float*     W   = (const float*)d_in[2];
    const float*     b   = (const float*)d_in[3];
    float*           out = (float*)d_out;

    const long long* src = ei;             // edge_index[0]
    const long long* dst = ei + N_EDGES;   // edge_index[1]

    // workspace layout: dinv | h1 | h2
    float* dinv = (float*)d_ws;
    float* h1   = (float*)((char*)d_ws + ((N_NODES * 4 + 255) & ~255ULL));
    float* h2   = h1 + (size_t)N_NODES * D_FEAT;

    // normalization
    k_deg_init <<<(N_NODES + 255) / 256, 256, 0, stream>>>(dinv);
    k_deg_accum<<<(N_EDGES + 255) / 256, 256, 0, stream>>>(dst, dinv);
    k_dinv     <<<(N_NODES + 255) / 256, 256, 0, stream>>>(dinv);

    // hop 1: x -> h1
    {
        long long t = (long long)N_NODES * (D_FEAT / 4);
        k_self<<<(unsigned)((t + 255) / 256), 256, 0, stream>>>(x, dinv, h1);
        k_edge<<<(N_EDGES + 7) / 8, 256, 0, stream>>>(src, dst, dinv, x, h1);
    }
    // hop 2: h1 -> h2
    {
        long long t = (long long)N_NODES * (D_FEAT / 4);
        k_self<<<(unsigned)((t + 255) / 256), 256, 0, stream>>>(h1, dinv, h2);
        k_edge<<<(N_EDGES + 7) / 8, 256, 0, stream>>>(src, dst, dinv, h1, h2);
    }

    // classifier + log-softmax (WMMA f32)
    k_logits<<<(N_TILES + 7) / 8, 256, 0, stream>>>(h2, W, b, out);
}